// LSTM_base_49014166782178
// MI455X (gfx1250) — compile-verified
//
#include <hip/hip_runtime.h>
#include <math.h>

// ---------------------------------------------------------------------------
// Problem dims (match reference)
// ---------------------------------------------------------------------------
#define B_      64
#define V_      50
#define S_      40
#define D_      512
#define H_      512
#define G4_     2048   // 4*H
#define TF_     64
#define NWG_LSTM 16
#define TPB      256   // 8 waves (wave32)

typedef float v2f __attribute__((ext_vector_type(2)));
typedef float v8f __attribute__((ext_vector_type(8)));

// fp32 WMMA: D = A(16x4) * B(4x16) + C, full fp32 path (matches fp32 reference)
__device__ __forceinline__ v8f wmma_f32_k4(v2f a, v2f b, v8f c) {
  return __builtin_amdgcn_wmma_f32_16x16x4_f32(false, a, false, b, (short)0, c,
                                               false, false);
}

__device__ __forceinline__ float sigm(float x) {
  return 1.0f / (1.0f + expf(-x));
}

// ---------------------------------------------------------------------------
// Kernel 1: ve[b,v,:] = sum_s relu(emb[idx]) + timeMLP(seq_time_step[b,v])
// ---------------------------------------------------------------------------
__global__ __launch_bounds__(TPB) void embed_time_kernel(
    const float* __restrict__ seq_time, const float* __restrict__ emb,
    const int* __restrict__ seqs, const float* __restrict__ tW1,
    const float* __restrict__ tb1, const float* __restrict__ tW2,
    const float* __restrict__ tb2, float* __restrict__ ve) {
  __shared__ float tf_s[TF_];
  __shared__ int idx_s[S_];
  const int bv = blockIdx.x;  // b*V + v
  const int tid = threadIdx.x;
  if (tid < TF_) {
    float t = seq_time[bv] * (1.0f / 180.0f);
    float z = t * tW1[tid] + tb1[tid];
    tf_s[tid] = 1.0f - tanhf(z * z);
  }
  if (tid < S_) idx_s[tid] = seqs[bv * S_ + tid];
  __syncthreads();
  for (int d = tid; d < D_; d += TPB) {
    float acc = tb2[d];
#pragma unroll
    for (int j = 0; j < TF_; ++j) acc += tf_s[j] * tW2[d * TF_ + j];
    for (int s = 0; s < S_; ++s) {
      float v = emb[(size_t)idx_s[s] * D_ + d];  // rows coalesced across d
      acc += fmaxf(v, 0.0f);
    }
    ve[(size_t)bv * D_ + d] = acc;
  }
}

// ---------------------------------------------------------------------------
// Kernel 2: bsum = bih + bhh (folded into the precomputed input gates)
// ---------------------------------------------------------------------------
__global__ void bias_sum_kernel(const float* __restrict__ a,
                                const float* __restrict__ b,
                                float* __restrict__ out, int n) {
  const int i = blockIdx.x * blockDim.x + threadIdx.x;
  if (i < n) out[i] = a[i] + b[i];
}

// ---------------------------------------------------------------------------
// Kernel 3: fp32 WMMA GEMM, 4 N-subtiles per wave (A-fragment reuse x4),
// compile-time K, manual 4-chunk rotating prefetch (20 b64 loads in flight).
//   C[M x N] = act(A[M x K] @ W[N x K]^T + bias)
// ---------------------------------------------------------------------------
template <int K, int RELU>
__global__ __launch_bounds__(TPB) void gemm4_kernel(
    const float* __restrict__ A, const float* __restrict__ W,
    const float* __restrict__ bias, float* __restrict__ C, int M, int N) {
  constexpr int NC = K / 4;  // K-chunks
  constexpr int PF = 4;      // prefetch distance (chunks)
  static_assert((NC - PF) % 4 == 0, "unroll grouping keeps slot index static");

  const int gid = blockIdx.x * (blockDim.x >> 5) + (threadIdx.x >> 5);
  const int mtiles = M >> 4;
  const int ngroups = N >> 6;
  if (gid >= mtiles * ngroups) return;  // wave-uniform
  const int mt = gid % mtiles;
  const int ng = gid / mtiles;
  const int lane = threadIdx.x & 31;
  const int lm = lane & 15;
  const int half = lane >> 4;
  const int koff = half * 2;

  const float* ar = A + (size_t)(mt * 16 + lm) * K + koff;
  const float* w0 = W + (size_t)(ng * 64 + 0 * 16 + lm) * K + koff;
  const float* w1 = W + (size_t)(ng * 64 + 1 * 16 + lm) * K + koff;
  const float* w2 = W + (size_t)(ng * 64 + 2 * 16 + lm) * K + koff;
  const float* w3 = W + (size_t)(ng * 64 + 3 * 16 + lm) * K + koff;

  v2f ab[PF], bb0[PF], bb1[PF], bb2[PF], bb3[PF];
#pragma unroll
  for (int p = 0; p < PF; ++p) {
    ab[p]  = *(const v2f*)(ar + p * 4);
    bb0[p] = *(const v2f*)(w0 + p * 4);
    bb1[p] = *(const v2f*)(w1 + p * 4);
    bb2[p] = *(const v2f*)(w2 + p * 4);
    bb3[p] = *(const v2f*)(w3 + p * 4);
  }

  v8f a0 = {}, a1 = {}, a2 = {}, a3 = {};
#pragma unroll 4
  for (int c = 0; c < NC - PF; ++c) {
    const int s = c & (PF - 1);
    const v2f av = ab[s];
    const v2f v0 = bb0[s], v1 = bb1[s], v2 = bb2[s], v3 = bb3[s];
    const int kn = (c + PF) * 4;
    ab[s]  = *(const v2f*)(ar + kn);
    bb0[s] = *(const v2f*)(w0 + kn);
    bb1[s] = *(const v2f*)(w1 + kn);
    bb2[s] = *(const v2f*)(w2 + kn);
    bb3[s] = *(const v2f*)(w3 + kn);
    a0 = wmma_f32_k4(av, v0, a0);
    a1 = wmma_f32_k4(av, v1, a1);
    a2 = wmma_f32_k4(av, v2, a2);
    a3 = wmma_f32_k4(av, v3, a3);
  }
#pragma unroll
  for (int c = NC - PF; c < NC; ++c) {
    const int s = c & (PF - 1);
    a0 = wmma_f32_k4(ab[s], bb0[s], a0);
    a1 = wmma_f32_k4(ab[s], bb1[s], a1);
    a2 = wmma_f32_k4(ab[s], bb2[s], a2);
    a3 = wmma_f32_k4(ab[s], bb3[s], a3);
  }

#pragma unroll
  for (int s = 0; s < 4; ++s) {
    const v8f acc = (s == 0) ? a0 : (s == 1) ? a1 : (s == 2) ? a2 : a3;
    const int n = ng * 64 + s * 16 + lm;
    const float b = bias[n];
#pragma unroll
    for (int r = 0; r < 8; ++r) {
      const int m = mt * 16 + r + 8 * half;
      float v = acc[r] + b;
      if (RELU) v = fmaxf(v, 0.0f);
      C[(size_t)m * N + n] = v;
    }
  }
}

// ---------------------------------------------------------------------------
// Grid barrier for the persistent recurrence kernel (16 co-resident WGs)
// ---------------------------------------------------------------------------
__device__ __forceinline__ void grid_barrier(int* counter, int& round) {
  __syncthreads();
  __threadfence();
  ++round;
  if (threadIdx.x == 0) {
    atomicAdd(counter, 1);
    const int target = NWG_LSTM * round;
    while (__hip_atomic_load(counter, __ATOMIC_RELAXED,
                             __HIP_MEMORY_SCOPE_AGENT) < target) {
      __builtin_amdgcn_s_sleep(2);
    }
  }
  __syncthreads();
  __threadfence();
}

// ---------------------------------------------------------------------------
// Kernel 4: persistent LSTM recurrence. Input gates Gx precomputed; per step
// only h @ Whh^T remains. Whh fragments staged once in LDS (256KB dynamic):
// layout [(wtLocal*4+gate)*128 + chunk][lane*2] -> conflict-free ds_load_b64.
// Gx gate values and c are prefetched at the top of each step so their global
// latency hides behind the 512 WMMAs of the h-GEMM.
// ---------------------------------------------------------------------------
__global__ __launch_bounds__(TPB) void lstm_rec_kernel(
    const float* __restrict__ Gx, const float* __restrict__ Whh,
    float* __restrict__ h0buf, float* __restrict__ h1buf,
    float* __restrict__ cbuf, int* __restrict__ counter) {
  extern __shared__ float wlds[];  // 8 * 128 * 64 floats = 256 KB
  constexpr int NC = H_ / 4;  // 128 chunks
  constexpr int PF = 4;
  const int waveId = threadIdx.x >> 5;
  const int lane = threadIdx.x & 31;
  const int lm = lane & 15;
  const int half = lane >> 4;
  const int koff = half * 2;

  const int mt = waveId & 3;                 // batch tile 0..3
  const int wtLocal = waveId >> 2;           // 0..1
  const int wt = blockIdx.x * 2 + wtLocal;   // hidden tile 0..31
  const int j = wt * 16 + lm;                // hidden column (B frag)

  // --- One-time stage of Whh fragments into LDS.
  {
    const int g = waveId & 3;
    const int wl = waveId >> 2;
    const float* src =
        Whh + (size_t)(g * H_ + (blockIdx.x * 2 + wl) * 16 + lm) * H_ + koff;
    float* dst = &wlds[(size_t)waveId * 128 * 64 + lane * 2];
#pragma unroll 8
    for (int c = 0; c < 128; ++c) {
      *(v2f*)(dst + c * 64) = *(const v2f*)(src + c * 4);
    }
  }

  // h(0) = c(0) = 0 for the cells this wave owns (unique ownership).
#pragma unroll
  for (int r = 0; r < 8; ++r) {
    const int m = mt * 16 + r + 8 * half;
    h0buf[m * H_ + j] = 0.0f;
    cbuf[m * H_ + j] = 0.0f;
  }
  __syncthreads();
  int round = 0;
  grid_barrier(counter, round);

  const float* wf0 = &wlds[(size_t)(wtLocal * 4 + 0) * 128 * 64 + lane * 2];
  const float* wf1 = &wlds[(size_t)(wtLocal * 4 + 1) * 128 * 64 + lane * 2];
  const float* wf2 = &wlds[(size_t)(wtLocal * 4 + 2) * 128 * 64 + lane * 2];
  const float* wf3 = &wlds[(size_t)(wtLocal * 4 + 3) * 128 * 64 + lane * 2];

  for (int t = 0; t < V_; ++t) {
    const float* hcur = (t & 1) ? h1buf : h0buf;
    float* hnext = (t & 1) ? h0buf : h1buf;
    const float* hrow = hcur + (size_t)(mt * 16 + lm) * H_ + koff;

    // --- Early issue: this step's input-gate values and c state (40 loads,
    // latency hidden behind the WMMA loop below).
    float gx0[8], gx1[8], gx2[8], gx3[8], cp[8];
#pragma unroll
    for (int r = 0; r < 8; ++r) {
      const int m = mt * 16 + r + 8 * half;
      const size_t grow = ((size_t)m * V_ + t) * G4_;
      gx0[r] = Gx[grow + 0 * H_ + j];
      gx1[r] = Gx[grow + 1 * H_ + j];
      gx2[r] = Gx[grow + 2 * H_ + j];
      gx3[r] = Gx[grow + 3 * H_ + j];
      cp[r] = cbuf[m * H_ + j];
    }

    // --- h @ Whh^T with rotating prefetch (A: global, B: LDS)
    v2f ab[PF], bb0[PF], bb1[PF], bb2[PF], bb3[PF];
#pragma unroll
    for (int p = 0; p < PF; ++p) {
      ab[p]  = *(const v2f*)(hrow + p * 4);
      bb0[p] = *(const v2f*)(wf0 + p * 64);
      bb1[p] = *(const v2f*)(wf1 + p * 64);
      bb2[p] = *(const v2f*)(wf2 + p * 64);
      bb3[p] = *(const v2f*)(wf3 + p * 64);
    }
    v8f ai = {}, af = {}, ag = {}, ao = {};
#pragma unroll 4
    for (int c = 0; c < NC - PF; ++c) {
      const int s = c & (PF - 1);
      const v2f av = ab[s];
      const v2f v0 = bb0[s], v1 = bb1[s], v2 = bb2[s], v3 = bb3[s];
      ab[s]  = *(const v2f*)(hrow + (c + PF) * 4);
      bb0[s] = *(const v2f*)(wf0 + (c + PF) * 64);
      bb1[s] = *(const v2f*)(wf1 + (c + PF) * 64);
      bb2[s] = *(const v2f*)(wf2 + (c + PF) * 64);
      bb3[s] = *(const v2f*)(wf3 + (c + PF) * 64);
      ai = wmma_f32_k4(av, v0, ai);
      af = wmma_f32_k4(av, v1, af);
      ag = wmma_f32_k4(av, v2, ag);
      ao = wmma_f32_k4(av, v3, ao);
    }
#pragma unroll
    for (int c = NC - PF; c < NC; ++c) {
      const int s = c & (PF - 1);
      ai = wmma_f32_k4(ab[s], bb0[s], ai);
      af = wmma_f32_k4(ab[s], bb1[s], af);
      ag = wmma_f32_k4(ab[s], bb2[s], ag);
      ao = wmma_f32_k4(ab[s], bb3[s], ao);
    }

    // --- Elementwise LSTM update; Gx already holds x@Wih^T + bih + bhh.
    // Accumulator element r is (m = mt*16 + r + 8*half, n = lane&15).
#pragma unroll
    for (int r = 0; r < 8; ++r) {
      const int m = mt * 16 + r + 8 * half;
      const float ig = sigm(ai[r] + gx0[r]);
      const float fg = sigm(af[r] + gx1[r]);
      const float gg = tanhf(ag[r] + gx2[r]);
      const float og = sigm(ao[r] + gx3[r]);
      const float cn = fg * cp[r] + ig * gg;
      cbuf[m * H_ + j] = cn;
      hnext[m * H_ + j] = og * tanhf(cn);
    }
    grid_barrier(counter, round);
  }
}

// ---------------------------------------------------------------------------
// Kernel 5: final projection  logits[64,2] = h2 @ cW3^T + cb3
// ---------------------------------------------------------------------------
__global__ void final_proj_kernel(const float* __restrict__ h2,
                                  const float* __restrict__ cW3,
                                  const float* __restrict__ cb3,
                                  float* __restrict__ out) {
  const int i = blockIdx.x * blockDim.x + threadIdx.x;  // 0..127
  if (i < B_ * 2) {
    const int m = i >> 1;
    const int n = i & 1;
    float acc = cb3[n];
    for (int k = 0; k < 2 * D_; ++k)
      acc += h2[(size_t)m * (2 * D_) + k] * cW3[(size_t)n * (2 * D_) + k];
    out[i] = acc;
  }
}

// ---------------------------------------------------------------------------
// Launch
// ---------------------------------------------------------------------------
extern "C" void kernel_launch(void* const* d_in, const int* in_sizes, int n_in,
                              void* d_out, int out_size, void* d_ws,
                              size_t ws_size, hipStream_t stream) {
  (void)in_sizes; (void)n_in; (void)out_size; (void)ws_size;
  const float* seq_time = (const float*)d_in[0];
  const float* emb      = (const float*)d_in[1];
  const float* tW1      = (const float*)d_in[2];
  const float* tb1      = (const float*)d_in[3];
  const float* tW2      = (const float*)d_in[4];
  const float* tb2      = (const float*)d_in[5];
  const float* Wih      = (const float*)d_in[6];
  const float* Whh      = (const float*)d_in[7];
  const float* bih      = (const float*)d_in[8];
  const float* bhh      = (const float*)d_in[9];
  const float* cW1      = (const float*)d_in[10];
  const float* cb1      = (const float*)d_in[11];
  const float* cW2      = (const float*)d_in[12];
  const float* cb2      = (const float*)d_in[13];
  const float* cW3      = (const float*)d_in[14];
  const float* cb3      = (const float*)d_in[15];
  const int*   seqs     = (const int*)d_in[16];

  float* ws = (float*)d_ws;
  size_t off = 0;
  float* ve    = ws + off; off += (size_t)B_ * V_ * D_;    // 1,638,400 f
  float* Gx    = ws + off; off += (size_t)B_ * V_ * G4_;   // 6,553,600 f
  float* h0buf = ws + off; off += (size_t)B_ * H_;
  float* h1buf = ws + off; off += (size_t)B_ * H_;
  float* cbuf  = ws + off; off += (size_t)B_ * H_;
  float* bsum  = ws + off; off += (size_t)G4_;
  float* hl1   = ws + off; off += (size_t)B_ * 4 * D_;
  float* hl2   = ws + off; off += (size_t)B_ * 2 * D_;
  int* counter = (int*)(ws + off);

  hipMemsetAsync(counter, 0, sizeof(int), stream);

  // Stage 1: visit embeddings + time features -> ve[3200, 512]
  embed_time_kernel<<<B_ * V_, TPB, 0, stream>>>(seq_time, emb, seqs, tW1, tb1,
                                                 tW2, tb2, ve);
  // Stage 2a: combined bias
  bias_sum_kernel<<<(G4_ + TPB - 1) / TPB, TPB, 0, stream>>>(bih, bhh, bsum,
                                                             G4_);
  // Stage 2b: precompute all input gates off the critical path:
  // Gx[3200, 2048] = ve @ Wih^T + (bih + bhh). 200 mtiles x 32 ngroups = 6400
  // wave-tasks -> 800 WGs.
  gemm4_kernel<D_, 0><<<800, TPB, 0, stream>>>(ve, Wih, bsum, Gx, B_ * V_,
                                               G4_);
  // Stage 3: persistent 50-step recurrence (h @ Whh^T only), Whh in LDS.
  // Final h lands in h0buf (t=49 writes buf[(49+1)&1] = buf0).
  lstm_rec_kernel<<<NWG_LSTM, TPB, 8 * 128 * 64 * sizeof(float), stream>>>(
      Gx, Whh, h0buf, h1buf, cbuf, counter);
  // Stage 4: classifier on final hidden state only (logits[:, -1, :])
  gemm4_kernel<D_, 1><<<16, TPB, 0, stream>>>(h0buf, cW1, cb1, hl1, B_,
                                              4 * D_);
  gemm4_kernel<4 * D_, 1><<<8, TPB, 0, stream>>>(hl1, cW2, cb2, hl2, B_,
                                                 2 * D_);
  final_proj_kernel<<<1, 128, 0, stream>>>(hl2, cW3, cb3, (float*)d_out);
}